// Wrapper_67018669687581
// MI455X (gfx1250) — compile-verified
//
#include <hip/hip_runtime.h>
#include <stdint.h>

#define MAX_DET 300
#define CONF_THR 0.5f
#define IOU_THR 0.4f
#define NMS_EPS 1e-9f
#define NCLS 80
#define TILE 128
#define CAP 1024

typedef unsigned int uint32;

// ---- sortable key transform (monotonic float -> uint) ----
__device__ __forceinline__ uint32 enc_key(float f) {
  uint32 b = __float_as_uint(f);
  return (b & 0x80000000u) ? ~b : (b | 0x80000000u);
}
__device__ __forceinline__ float dec_key(uint32 k) {
  uint32 b = (k & 0x80000000u) ? (k & 0x7FFFFFFFu) : ~k;
  return __uint_as_float(b);
}

#if __has_builtin(__builtin_amdgcn_tensor_load_to_lds) && __has_builtin(__builtin_amdgcn_s_wait_tensorcnt)
#define USE_TDM 1
#endif

typedef unsigned int v4u __attribute__((ext_vector_type(4)));
typedef int v4i __attribute__((ext_vector_type(4)));
typedef int v8i __attribute__((ext_vector_type(8)));

// ---------------------------------------------------------------------------
// Kernel 1: per-anchor max/argmax over 80 class rows.
// TDM streams an (80 x TILE) fp32 tile of the class-score block into LDS;
// waves reduce from LDS. Bandwidth-bound: one 86 MB streaming read.
// ---------------------------------------------------------------------------
__global__ void score_reduce_kernel(const float* __restrict__ pred,
                                    uint32* __restrict__ keys,
                                    int* __restrict__ cls,
                                    int N) {
  const int t = threadIdx.x;
  const int blk = blockIdx.x;
  const int a = blk * TILE + t;
#ifdef USE_TDM
  __shared__ float tile[NCLS * TILE]; // 40 KB
  if (t < 32) { // wave 0 issues the tensor DMA (EXEC ignored by TDM)
    unsigned long long ga = (unsigned long long)(uintptr_t)pred +
                            4ull * (4ull * (unsigned long long)N +
                                    (unsigned long long)blk * (unsigned long long)TILE);
    uint32 ldsoff = (uint32)(uintptr_t)(&tile[0]);
    // D# group 0: count=1 (valid), lds_addr, global_addr[56:0], type=2
    v4u g0;
    g0[0] = 1u;
    g0[1] = ldsoff;
    g0[2] = (uint32)(ga & 0xFFFFFFFFull);
    g0[3] = (uint32)((ga >> 32) & 0x1FFFFFFull) | (2u << 30);
    // D# group 1: data_size=4B, tensor_dim0=N, tensor_dim1=80,
    //             tile_dim0=TILE, tile_dim1=80, stride0=N
    v8i g1;
    g1[0] = (int)(2u << 16);                                  // data_size=2 (4B)
    g1[1] = (int)(((uint32)N & 0xFFFFu) << 16);               // tensor_dim0[15:0]
    g1[2] = (int)(((uint32)N >> 16) | ((uint32)NCLS << 16));  // td0[31:16] | td1[15:0]
    g1[3] = (int)((uint32)TILE << 16);                        // td1[31:16]=0 | tile_dim0
    g1[4] = (int)(uint32)NCLS;                                // tile_dim1 | tile_dim2=0
    g1[5] = (int)(uint32)N;                                   // stride0[31:0]
    g1[6] = 0;                                                // stride0[47:32]|stride1 lo
    g1[7] = 0;
    v4i g2 = {0, 0, 0, 0};
    v4i g3 = {0, 0, 0, 0};
#if defined(__clang_major__) && (__clang_major__ >= 23)
    v8i g4 = {0, 0, 0, 0, 0, 0, 0, 0};
    __builtin_amdgcn_tensor_load_to_lds(g0, g1, g2, g3, g4, 0);
#else
    __builtin_amdgcn_tensor_load_to_lds(g0, g1, g2, g3, 0);
#endif
    __builtin_amdgcn_s_wait_tensorcnt(0);
  }
  __syncthreads();
  float best = -1.0f;
  int bc = 0;
#pragma unroll 8
  for (int c = 0; c < NCLS; ++c) {
    float v = tile[c * TILE + t];
    if (v > best) { best = v; bc = c; }
  }
#else
  float best = -1.0f;
  int bc = 0;
  if (a < N) {
    for (int c = 0; c < NCLS; ++c) {
      float v = pred[(size_t)(4 + c) * (size_t)N + a];
      if (v > best) { best = v; bc = c; }
    }
  }
#endif
  if (a < N) {
    keys[a] = enc_key(best);
    cls[a] = bc;
  }
}

// ---------------------------------------------------------------------------
// Radix-select scratch init (every call: ws is not re-zeroed by harness)
// ---------------------------------------------------------------------------
__global__ void init_kernel(uint32* hist, uint32* state, uint32* counter) {
  int t = threadIdx.x;
  hist[t] = 0;
  if (t == 0) {
    state[0] = 0;        // accumulated high-bit prefix
    state[1] = MAX_DET;  // remaining rank
    counter[0] = 0;      // candidate append counter
  }
}

// ---------------------------------------------------------------------------
// Radix select pass: 256-bin histogram of byte (24 - 8*pass), filtered on the
// already-fixed high bytes.
// ---------------------------------------------------------------------------
__global__ void hist_kernel(const uint32* __restrict__ keys,
                            uint32* __restrict__ hist,
                            const uint32* __restrict__ state,
                            int N, int pass) {
  __shared__ uint32 lh[256];
  int t = threadIdx.x;
  lh[t] = 0;
  __syncthreads();
  int a = blockIdx.x * 256 + t;
  if (a < N) {
    uint32 k = keys[a];
    uint32 hi = (pass == 0) ? 0u : (k >> (32 - 8 * pass));
    uint32 prefix = (pass == 0) ? 0u : state[0];
    if (hi == prefix) {
      uint32 bin = (k >> (24 - 8 * pass)) & 0xFFu;
      atomicAdd(&lh[bin], 1u);
    }
  }
  __syncthreads();
  if (lh[t]) atomicAdd(&hist[t], lh[t]);
}

// Single-thread scan: pick the bin where cumulative (from high) crosses the
// remaining rank; extend prefix; zero histogram for the next pass.
__global__ void scan_kernel(uint32* hist, uint32* state) {
  uint32 K = state[1];
  uint32 cum = 0;
  uint32 bsel = 0;
  for (int b = 255; b >= 0; --b) {
    uint32 h = hist[b];
    if (cum + h >= K) { bsel = (uint32)b; K = K - cum; break; }
    cum += h;
  }
  state[0] = (state[0] << 8) | bsel;
  state[1] = K;
  for (int b = 0; b < 256; ++b) hist[b] = 0;
}

// ---------------------------------------------------------------------------
// Gather every key >= exact 300th-largest key (>= 300 entries, ~300 expected)
// ---------------------------------------------------------------------------
__global__ void gather_kernel(const uint32* __restrict__ keys,
                              const uint32* __restrict__ state,
                              uint32* counter,
                              uint32* cand_key, int* cand_idx, int N) {
  int a = blockIdx.x * 256 + threadIdx.x;
  if (a + 8192 < N) __builtin_prefetch(&keys[a + 8192], 0, 0);
  if (a < N) {
    uint32 k = keys[a];
    if (k >= state[0]) {
      uint32 pos = atomicAdd(counter, 1u);
      if (pos < CAP) {
        cand_key[pos] = k;
        cand_idx[pos] = a;
      }
    }
  }
}

// ---------------------------------------------------------------------------
// Single-block bitonic sort (desc key, asc index tie-break) -> top 300.
// Canonicalizes the nondeterministic atomic-append order.
// ---------------------------------------------------------------------------
__global__ void sort_kernel(const uint32* __restrict__ counter,
                            const uint32* __restrict__ cand_key,
                            const int* __restrict__ cand_idx,
                            uint32* __restrict__ top_key,
                            int* __restrict__ top_idx) {
  __shared__ uint32 sk[CAP];
  __shared__ int si[CAP];
  int t = threadIdx.x;
  uint32 M = counter[0];
  if (M > CAP) M = CAP;
  if ((uint32)t < M) { sk[t] = cand_key[t]; si[t] = cand_idx[t]; }
  else               { sk[t] = 0u;          si[t] = 0x7FFFFFFF; }
  __syncthreads();
  for (int k = 2; k <= CAP; k <<= 1) {
    for (int j = k >> 1; j > 0; j >>= 1) {
      int ixj = t ^ j;
      if (ixj > t) {
        bool descBlock = ((t & k) == 0);
        uint32 ka = sk[t], kb = sk[ixj];
        int ia = si[t], ib = si[ixj];
        bool before = (ka > kb) || (ka == kb && ia < ib);
        if (before != descBlock) {
          sk[t] = kb; si[t] = ib;
          sk[ixj] = ka; si[ixj] = ia;
        }
      }
      __syncthreads();
    }
  }
  if (t < MAX_DET) { top_key[t] = sk[t]; top_idx[t] = si[t]; }
}

// ---------------------------------------------------------------------------
// Single-block greedy NMS (serial i-loop, matches reference fori_loop) + clip
// + masked 300x6 output.
// ---------------------------------------------------------------------------
__global__ void nms_out_kernel(const float* __restrict__ pred,
                               const int* __restrict__ cls,
                               const uint32* __restrict__ top_key,
                               const int* __restrict__ top_idx,
                               const int* __restrict__ ohp,
                               const int* __restrict__ owp,
                               float* __restrict__ out, int N) {
  __shared__ float X1[MAX_DET], Y1[MAX_DET], X2[MAX_DET], Y2[MAX_DET];
  __shared__ float AR[MAX_DET], SC[MAX_DET], CL[MAX_DET];
  __shared__ int KP[MAX_DET];
  int j = threadIdx.x;
  if (j < MAX_DET) {
    int i = top_idx[j];
    float cx = pred[(size_t)0 * N + i];
    float cy = pred[(size_t)1 * N + i];
    float w  = pred[(size_t)2 * N + i];
    float h  = pred[(size_t)3 * N + i];
    float sc = dec_key(top_key[j]);
    X1[j] = cx - 0.5f * w; Y1[j] = cy - 0.5f * h;
    X2[j] = cx + 0.5f * w; Y2[j] = cy + 0.5f * h;
    AR[j] = (X2[j] - X1[j]) * (Y2[j] - Y1[j]);
    SC[j] = sc;
    CL[j] = (float)cls[i];
    KP[j] = (sc > CONF_THR) ? 1 : 0;
  }
  for (int i = 0; i < MAX_DET; ++i) {
    __syncthreads();
    if (j < MAX_DET && j > i && KP[i] && KP[j]) {
      float iw = fmaxf(fminf(X2[i], X2[j]) - fmaxf(X1[i], X1[j]), 0.0f);
      float ih = fmaxf(fminf(Y2[i], Y2[j]) - fmaxf(Y1[i], Y1[j]), 0.0f);
      float inter = iw * ih;
      float iou = inter / (AR[i] + AR[j] - inter + NMS_EPS);
      if (iou > IOU_THR) KP[j] = 0;
    }
  }
  __syncthreads();
  if (j < MAX_DET) {
    float W = (float)(*owp), H = (float)(*ohp);
    float o0 = 0.f, o1 = 0.f, o2 = 0.f, o3 = 0.f, o4 = 0.f, o5 = 0.f;
    if (KP[j]) {
      o0 = fminf(fmaxf(X1[j], 0.0f), W);
      o1 = fminf(fmaxf(Y1[j], 0.0f), H);
      o2 = fminf(fmaxf(X2[j], 0.0f), W);
      o3 = fminf(fmaxf(Y2[j], 0.0f), H);
      o4 = SC[j];
      o5 = CL[j];
    }
    out[j * 6 + 0] = o0; out[j * 6 + 1] = o1; out[j * 6 + 2] = o2;
    out[j * 6 + 3] = o3; out[j * 6 + 4] = o4; out[j * 6 + 5] = o5;
  }
}

extern "C" void kernel_launch(void* const* d_in, const int* in_sizes, int n_in,
                              void* d_out, int out_size, void* d_ws, size_t ws_size,
                              hipStream_t stream) {
  const float* pred = (const float*)d_in[0];
  const int* ohp = (const int*)d_in[1];
  const int* owp = (const int*)d_in[2];
  float* out = (float*)d_out;
  const int N = in_sizes[0] / (NCLS + 4);

  // workspace carve-out (all 4-byte aligned types)
  char* ws = (char*)d_ws;
  uint32* keys     = (uint32*)ws; ws += (size_t)N * 4;
  int*    cls      = (int*)ws;    ws += (size_t)N * 4;
  uint32* hist     = (uint32*)ws; ws += 256 * 4;
  uint32* state    = (uint32*)ws; ws += 16 * 4;
  uint32* counter  = (uint32*)ws; ws += 16 * 4;
  uint32* cand_key = (uint32*)ws; ws += CAP * 4;
  int*    cand_idx = (int*)ws;    ws += CAP * 4;
  uint32* top_key  = (uint32*)ws; ws += ((MAX_DET + 3) / 4) * 16;
  int*    top_idx  = (int*)ws;    ws += ((MAX_DET + 3) / 4) * 16;

  const int gridR = (N + TILE - 1) / TILE;
  const int grid256 = (N + 255) / 256;

  score_reduce_kernel<<<gridR, TILE, 0, stream>>>(pred, keys, cls, N);
  init_kernel<<<1, 256, 0, stream>>>(hist, state, counter);
  for (int p = 0; p < 4; ++p) {
    hist_kernel<<<grid256, 256, 0, stream>>>(keys, hist, state, N, p);
    scan_kernel<<<1, 1, 0, stream>>>(hist, state);
  }
  gather_kernel<<<grid256, 256, 0, stream>>>(keys, state, counter, cand_key, cand_idx, N);
  sort_kernel<<<1, CAP, 0, stream>>>(counter, cand_key, cand_idx, top_key, top_idx);
  nms_out_kernel<<<1, 320, 0, stream>>>(pred, cls, top_key, top_idx, ohp, owp, out, N);
}